// ArcTransformer_14044543058154
// MI455X (gfx1250) — compile-verified
//
#include <hip/hip_runtime.h>
#include <hip/hip_bf16.h>

typedef float v2f __attribute__((ext_vector_type(2)));
typedef float v8f __attribute__((ext_vector_type(8)));

#define B_   2
#define T_   2048
#define NH_  8
#define DH_  64
#define D_   512
#define P_   16
#define NTOK (B_*T_)      /* 4096 token rows   */
#define NROW (NTOK*NH_)   /* 32768 token-head rows */

__device__ __forceinline__ v8f wmma4(v2f a, v2f b, v8f c) {
  // D = A(16x4 f32) * B(4x16 f32) + C(16x16 f32), wave32
  return __builtin_amdgcn_wmma_f32_16x16x4_f32(false, a, false, b, (short)0, c, false, false);
}
__device__ __forceinline__ v8f vzero8() { v8f z; for (int i = 0; i < 8; ++i) z[i] = 0.f; return z; }

// ---------------------------------------------------------------------------
// Kernel 1: x0 = embed[ids]; hN = layernorm(x0)   (one token per block)
// ---------------------------------------------------------------------------
__global__ void k_embed_ln(const int* __restrict__ ids, const float* __restrict__ embed,
                           const float* __restrict__ g, const float* __restrict__ bta,
                           float* __restrict__ x0, float* __restrict__ hN) {
  __shared__ float red[8];
  const int tok = blockIdx.x, tid = threadIdx.x;
  const int id = ids[tok];
  const float* e = embed + (long)id * D_;
  float v0 = e[tid], v1 = e[tid + 256];
  float s = v0 + v1;
  for (int m = 1; m < 32; m <<= 1) s += __shfl_xor(s, m);
  if ((tid & 31) == 0) red[tid >> 5] = s;
  __syncthreads();
  float tot = 0.f; for (int i = 0; i < 8; ++i) tot += red[i];
  const float mu = tot * (1.f / 512.f);
  __syncthreads();
  const float d0 = v0 - mu, d1 = v1 - mu;
  float q = d0 * d0 + d1 * d1;
  for (int m = 1; m < 32; m <<= 1) q += __shfl_xor(q, m);
  if ((tid & 31) == 0) red[tid >> 5] = q;
  __syncthreads();
  float tv = 0.f; for (int i = 0; i < 8; ++i) tv += red[i];
  const float rs = rsqrtf(tv * (1.f / 512.f) + 1e-5f);
  const long base = (long)tok * D_;
  x0[base + tid] = v0; x0[base + tid + 256] = v1;
  hN[base + tid]       = d0 * rs * g[tid]       + bta[tid];
  hN[base + tid + 256] = d1 * rs * g[tid + 256] + bta[tid + 256];
}

// ---------------------------------------------------------------------------
// Kernel 2: fused MoE compose for q,k,v (experts shared, gates differ)
// 128 token-head rows per block, 16 rows per wave. X stays LDS-resident;
// S has its own LDS region; A-fragments are re-read from LDS (no big
// register arrays -> no scratch spills).
// ---------------------------------------------------------------------------
__global__ void k_compose(const float* __restrict__ hN,
                          const float* __restrict__ pq, const float* __restrict__ gq_,
                          const float* __restrict__ pk, const float* __restrict__ gk_,
                          const float* __restrict__ pv, const float* __restrict__ gv_,
                          const float* __restrict__ w1, const float* __restrict__ w2,
                          float* __restrict__ qO, float* __restrict__ kO, float* __restrict__ vO) {
  extern __shared__ float sm[];
  float* ldsX  = sm;                 // 128*66  X (resident whole kernel)
  float* ldsS  = ldsX + 128 * 66;    // 128*66  silu(H) staging (per-wave rows)
  float* ldsW1 = ldsS + 128 * 66;    // 64*66   (also holds 48x64 protos pre-loop)
  float* ldsW2 = ldsW1 + 64 * 66;    // 64*66
  float* ldsG  = ldsW2 + 64 * 66;    // 3*16*130 gate weights [compose][expert][row]

  const int tid = threadIdx.x;
  const int wave = tid >> 5, lane = tid & 31, half = lane >> 4, l15 = lane & 15;
  const int rowBase = blockIdx.x * 128;

  for (int i = tid; i < 128 * 64; i += 256) {
    int r = i >> 6, c = i & 63;
    ldsX[r * 66 + c] = hN[(long)(rowBase + r) * 64 + c];
  }
  // stage the three 16x64 proto matrices into ldsW1 rows [c3*16+p]
  for (int i = tid; i < 3 * 16 * 64; i += 256) {
    int r = i >> 6, c = i & 63;
    const float* src = (r < 16) ? pq : (r < 32) ? pk : pv;
    ldsW1[r * 66 + c] = src[(r & 15) * 64 + c];
  }
  __syncthreads();

  const int aRow = wave * 16 + l15;     // A-fragment row (X and S tiles)
  const int kOff = half * 2;            // per-lane K sub-offset

  // gate weights via WMMA: logits = X * proto^T, scale - gate, relu, eps-mask
  const float* gates[3] = {gq_, gk_, gv_};
  for (int c3 = 0; c3 < 3; ++c3) {
    v8f lg = vzero8();
    for (int k = 0; k < 16; ++k) {
      int kb = k * 4 + kOff;
      v2f a = *(const v2f*)&ldsX[aRow * 66 + kb];
      v2f b = *(const v2f*)&ldsW1[(c3 * 16 + l15) * 66 + kb];  // B[k][n]=proto[n][k]
      lg = wmma4(a, b, lg);
    }
    const float gg = gates[c3][l15];
    for (int r = 0; r < 8; ++r) {
      float rl = fmaxf(lg[r] * 0.125f - gg, 0.f);
      float wv = (rl > 1e-6f) ? rl : 0.f;
      ldsG[(c3 * 16 + l15) * 130 + (wave * 16 + r + 8 * half)] = wv;
    }
  }

  v8f aq[4], ak[4], av[4];
  for (int nt = 0; nt < 4; ++nt) { aq[nt] = vzero8(); ak[nt] = vzero8(); av[nt] = vzero8(); }

  for (int p = 0; p < P_; ++p) {
    __syncthreads();  // protect ldsW1/W2 from previous iteration readers
    const float* w1p = w1 + (long)p * 4096;
    const float* w2p = w2 + (long)p * 4096;
    for (int i = tid; i < 4096; i += 256) {
      int r = i >> 6, c = i & 63;
      ldsW1[r * 66 + c] = w1p[i];
      ldsW2[r * 66 + c] = w2p[i];
    }
    __syncthreads();

    // GEMM1: H = X * W1^T, silu, stage S into per-wave-private rows of ldsS
    for (int nt = 0; nt < 4; ++nt) {
      v8f hh = vzero8();
      for (int k = 0; k < 16; ++k) {
        int kb = k * 4 + kOff;
        v2f a = *(const v2f*)&ldsX[aRow * 66 + kb];
        v2f b = *(const v2f*)&ldsW1[(nt * 16 + l15) * 66 + kb];  // B[k][n]=w1[n][k]
        hh = wmma4(a, b, hh);
      }
      for (int r = 0; r < 8; ++r) {
        float x = hh[r];
        ldsS[(wave * 16 + r + 8 * half) * 66 + nt * 16 + l15] = x / (1.f + __expf(-x));
      }
    }
    // GEMM2: O = S * W2^T, accumulate with the three gate weights
    for (int nt = 0; nt < 4; ++nt) {
      v8f oo = vzero8();
      for (int k = 0; k < 16; ++k) {
        int kb = k * 4 + kOff;
        v2f a = *(const v2f*)&ldsS[aRow * 66 + kb];
        v2f b = *(const v2f*)&ldsW2[(nt * 16 + l15) * 66 + kb];
        oo = wmma4(a, b, oo);
      }
      for (int r = 0; r < 8; ++r) {
        const int m = wave * 16 + r + 8 * half;
        aq[nt][r] += oo[r] * ldsG[(0 * 16 + p) * 130 + m];
        ak[nt][r] += oo[r] * ldsG[(1 * 16 + p) * 130 + m];
        av[nt][r] += oo[r] * ldsG[(2 * 16 + p) * 130 + m];
      }
    }
  }
  for (int nt = 0; nt < 4; ++nt)
    for (int r = 0; r < 8; ++r) {
      long idx = (long)(rowBase + wave * 16 + r + 8 * half) * 64 + nt * 16 + l15;
      qO[idx] = aq[nt][r]; kO[idx] = ak[nt][r]; vO[idx] = av[nt][r];
    }
}

// ---------------------------------------------------------------------------
// Kernel 3: causal flash attention per (b, h, q-chunk of 128). No block
// barriers: each wave owns private LDS K/V/P tiles (causal trip counts differ).
// ---------------------------------------------------------------------------
__global__ void k_attn(const float* __restrict__ q, const float* __restrict__ kk,
                       const float* __restrict__ vv, float* __restrict__ attn) {
  extern __shared__ float sm[];
  const int tid = threadIdx.x, wave = tid >> 5, lane = tid & 31, half = lane >> 4, l15 = lane & 15;
  float* myK = sm + wave * 1056;              // 16*66
  float* myV = sm + 8 * 1056 + wave * 1056;   // 16*66
  float* myP = sm + 16 * 1056 + wave * 288;   // 16*18

  const int qc = blockIdx.x & 15;
  const int h  = (blockIdx.x >> 4) & 7;
  const int b  = blockIdx.x >> 7;
  const int qbase = qc * 128 + wave * 16;

  v2f qf[16];
  {
    const int qrow = qbase + l15;
    const float* qp = q + ((long)(b * T_ + qrow) * NH_ + h) * 64;
    for (int k = 0; k < 16; ++k) { int kb = k * 4 + half * 2; qf[k] = *(const v2f*)&qp[kb]; }
  }
  float mrun[8], lsum[8];
  for (int r = 0; r < 8; ++r) { mrun[r] = -3.4e38f; lsum[r] = 0.f; }
  v8f oa[4]; for (int nt = 0; nt < 4; ++nt) oa[nt] = vzero8();

  const int nT = qc * 8 + wave + 1;  // causal: kv tiles 0 .. qbase/16
  for (int t = 0; t < nT; ++t) {
    const int kvb = t * 16;
    for (int i = lane; i < 1024; i += 32) {
      int r = i >> 6, c = i & 63;
      long grow = ((long)(b * T_ + kvb + r) * NH_ + h) * 64;
      myK[r * 66 + c] = kk[grow + c];
      myV[r * 66 + c] = vv[grow + c];
    }
    v8f s = vzero8();
    for (int k = 0; k < 16; ++k) {
      int kb = k * 4 + half * 2;
      v2f bb = *(const v2f*)&myK[l15 * 66 + kb];  // B[k][n] = K[kvb+n][k]
      s = wmma4(qf[k], bb, s);
    }
    for (int r = 0; r < 8; ++r) {
      const int qrowR = qbase + r + 8 * half;
      const int kvcol = kvb + l15;
      float val = s[r] * 0.125f;
      val = (kvcol <= qrowR) ? val : -3.0e38f;
      float mt = val;
      mt = fmaxf(mt, __shfl_xor(mt, 1));
      mt = fmaxf(mt, __shfl_xor(mt, 2));
      mt = fmaxf(mt, __shfl_xor(mt, 4));
      mt = fmaxf(mt, __shfl_xor(mt, 8));
      const float mnew = fmaxf(mrun[r], mt);
      float pr = __expf(val - mnew);
      float rs = pr;
      rs += __shfl_xor(rs, 1); rs += __shfl_xor(rs, 2);
      rs += __shfl_xor(rs, 4); rs += __shfl_xor(rs, 8);
      const float alpha = __expf(mrun[r] - mnew);
      lsum[r] = lsum[r] * alpha + rs;
      mrun[r] = mnew;
      for (int nt = 0; nt < 4; ++nt) oa[nt][r] *= alpha;
      myP[(r + 8 * half) * 18 + l15] = pr;
    }
    for (int nt = 0; nt < 4; ++nt)
      for (int k4 = 0; k4 < 4; ++k4) {
        int kb = k4 * 4 + half * 2;
        v2f a = *(const v2f*)&myP[l15 * 18 + kb];           // A[m][k] = P[m][k]
        v2f bb;
        bb.x = myV[kb * 66 + nt * 16 + l15];                // B[k][n] = V[kvb+k][n]
        bb.y = myV[(kb + 1) * 66 + nt * 16 + l15];
        oa[nt] = wmma4(a, bb, oa[nt]);
      }
  }
  for (int nt = 0; nt < 4; ++nt)
    for (int r = 0; r < 8; ++r) {
      const int tok = qbase + r + 8 * half;
      attn[((long)(b * T_ + tok) * NH_ + h) * 64 + nt * 16 + l15] = oa[nt][r] / lsum[r];
    }
}

// ---------------------------------------------------------------------------
// Kernel 4: x1 = x0 + attn @ o_w^T   (M=4096, N=512, K=512)
// ---------------------------------------------------------------------------
__global__ void k_proj(const float* __restrict__ attn, const float* __restrict__ ow,
                       const float* __restrict__ x0, float* __restrict__ x1) {
  extern __shared__ float sm[];
  float* ldsA = sm;                // 128*66
  float* ldsW = sm + 128 * 66;     // 64*66
  const int tid = threadIdx.x, wave = tid >> 5, lane = tid & 31, half = lane >> 4, l15 = lane & 15;
  const int rowBase = blockIdx.x * 128;
  for (int nb = 0; nb < 8; ++nb) {
    v8f acc[4]; for (int nt = 0; nt < 4; ++nt) acc[nt] = vzero8();
    for (int kc = 0; kc < 8; ++kc) {
      __syncthreads();
      for (int i = tid; i < 8192; i += 256) {
        int r = i >> 6, c = i & 63;
        ldsA[r * 66 + c] = attn[(long)(rowBase + r) * 512 + kc * 64 + c];
      }
      for (int i = tid; i < 4096; i += 256) {
        int r = i >> 6, c = i & 63;
        ldsW[r * 66 + c] = ow[(long)(nb * 64 + r) * 512 + kc * 64 + c];  // B[k][n]=ow[n][k]
      }
      __syncthreads();
      for (int nt = 0; nt < 4; ++nt)
        for (int k = 0; k < 16; ++k) {
          int kb = k * 4 + half * 2;
          v2f a  = *(const v2f*)&ldsA[(wave * 16 + l15) * 66 + kb];
          v2f bb = *(const v2f*)&ldsW[(nt * 16 + l15) * 66 + kb];
          acc[nt] = wmma4(a, bb, acc[nt]);
        }
    }
    for (int nt = 0; nt < 4; ++nt)
      for (int r = 0; r < 8; ++r) {
        long idx = (long)(rowBase + wave * 16 + r + 8 * half) * 512 + nb * 64 + nt * 16 + l15;
        x1[idx] = x0[idx] + acc[nt][r];
      }
  }
}

// ---------------------------------------------------------------------------
// Kernel 5: logits = x1 @ head_w^T   (N = 16 vocab, single n-tile)
// ---------------------------------------------------------------------------
__global__ void k_head(const float* __restrict__ x1, const float* __restrict__ hw,
                       float* __restrict__ out) {
  const int tid = threadIdx.x, wave = tid >> 5, lane = tid & 31, half = lane >> 4, l15 = lane & 15;
  const int rowBase = blockIdx.x * 128;
  const float* xp = x1 + (long)(rowBase + wave * 16 + l15) * 512;
  const float* bp = hw + (long)l15 * 512;   // B[k][n] = head_w[n][k]
  v8f acc = vzero8();
  for (int k = 0; k < 128; ++k) {
    int kb = k * 4 + half * 2;
    v2f a = *(const v2f*)&xp[kb];
    v2f b = *(const v2f*)&bp[kb];
    acc = wmma4(a, b, acc);
  }
  for (int r = 0; r < 8; ++r)
    out[(long)(rowBase + wave * 16 + r + 8 * half) * 16 + l15] = acc[r];
}

// ---------------------------------------------------------------------------
extern "C" void kernel_launch(void* const* d_in, const int* in_sizes, int n_in,
                              void* d_out, int out_size, void* d_ws, size_t ws_size,
                              hipStream_t stream) {
  (void)in_sizes; (void)n_in; (void)out_size; (void)ws_size;
  const int*   ids   = (const int*)  d_in[0];
  const float* embed = (const float*)d_in[1];
  const float* lng   = (const float*)d_in[2];
  const float* lnb   = (const float*)d_in[3];
  const float* pq    = (const float*)d_in[4];
  const float* gq    = (const float*)d_in[5];
  const float* pk    = (const float*)d_in[6];
  const float* gk    = (const float*)d_in[7];
  const float* pv    = (const float*)d_in[8];
  const float* gv    = (const float*)d_in[9];
  const float* w1    = (const float*)d_in[10];
  const float* w2    = (const float*)d_in[11];
  const float* ow    = (const float*)d_in[12];
  const float* hw    = (const float*)d_in[13];
  float* out = (float*)d_out;

  float* ws = (float*)d_ws;
  const size_t SZ = (size_t)NTOK * D_;   // 2,097,152 floats per buffer
  float* x0 = ws;
  float* hN = ws + SZ;
  float* qB = ws + 2 * SZ;
  float* kB = ws + 3 * SZ;
  float* vB = ws + 4 * SZ;
  float* at = ws + 5 * SZ;
  float* x1 = ws + 6 * SZ;

  k_embed_ln<<<NTOK, 256, 0, stream>>>(ids, embed, lng, lnb, x0, hN);

  const size_t smem2 = (size_t)(128 * 66 + 128 * 66 + 64 * 66 + 64 * 66 + 3 * 16 * 130) * sizeof(float);
  k_compose<<<NROW / 128, 256, smem2, stream>>>(hN, pq, gq, pk, gk, pv, gv, w1, w2, qB, kB, vB);

  const size_t smem3 = (size_t)(16 * 1056 + 8 * 288) * sizeof(float);
  k_attn<<<B_ * NH_ * (T_ / 128), 256, smem3, stream>>>(qB, kB, vB, at);

  const size_t smem4 = (size_t)(128 * 66 + 64 * 66) * sizeof(float);
  k_proj<<<NTOK / 128, 256, smem4, stream>>>(at, ow, x0, x1);

  k_head<<<NTOK / 128, 256, 0, stream>>>(x1, hw, out);
}